// GCN_new_77833397338523
// MI455X (gfx1250) — compile-verified
//
#include <hip/hip_runtime.h>

// ---------------------------------------------------------------------------
// GCN layer for MI455X (gfx1250, wave32):
//   h    = prelu(AX @ Wr^T + Wr_b, g_alpha)        (WMMA f32 GEMM)
//   temp = segment_sum(A_val * h[A_col], A_row)    (L2-resident f32 atomics)
//   out  = prelu(temp @ W^T + W_b, act_alpha)      (WMMA f32 GEMM)
// h and temp (51.2 MB each) both fit in the 192 MB L2, so the sparse
// gather/scatter phase runs at L2 bandwidth, not HBM. The GEMMs run on the
// matrix pipe via V_WMMA_F32_16X16X4_F32 (full fp32 fidelity).
// ---------------------------------------------------------------------------

typedef float v2f __attribute__((ext_vector_type(2)));
typedef float v8f __attribute__((ext_vector_type(8)));

#define DIM 128

// ----------------------------- GEMM + PReLU --------------------------------
// Y[m][n] = prelu( sum_k X[m][k] * W[n][k] + bias[n] )
// Block: 256 threads = 8 waves. Each wave computes a 16-row x 128-col strip.
// W is staged in LDS K-pair interleaved:
//   Wp[(k>>1)*128 + n] = { W[n][k], W[n][k+1] }   (k even)
// so a B fragment for WMMA is a single aligned ds_load_b64 into an even
// VGPR pair (WMMA SRC1 must be even-aligned).
__global__ __launch_bounds__(256) void gemm_prelu_wmma(
    const float* __restrict__ X,      // [N, 128]
    const float* __restrict__ W,      // [128, 128] row-major
    const float* __restrict__ bias,   // [128]
    const float* __restrict__ alpha_p,// [1]
    float* __restrict__ Y,            // [N, 128]
    int N)
{
    __shared__ v2f Wp[64 * DIM];      // 64 KB

    // Cooperative staging: Wp[kpair*128 + n] = (v2f)W[n][2*kpair .. 2*kpair+1]
    {
        const v2f* Wv = (const v2f*)W;          // Wv[n*64 + kpair]
        for (int i = threadIdx.x; i < 64 * DIM; i += 256) {
            const int kpair = i >> 7;           // 0..63
            const int n     = i & 127;          // 0..127
            Wp[i] = Wv[n * 64 + kpair];
        }
    }
    __syncthreads();

    const int lane    = threadIdx.x & 31;
    const int wave    = threadIdx.x >> 5;
    const int l15     = lane & 15;
    const int halfSel = lane >> 4;                 // 0: lanes 0-15, 1: lanes 16-31
    const int rowBase = (blockIdx.x * 8 + wave) * 16;

    // A-fragment source row (ISA 7.12.2: lanes 0-15 carry K={0,1}, lanes 16-31
    // carry K={2,3} of the same M=0..15 rows). Clamp so EXEC stays all-ones.
    int arow = rowBase + l15;
    if (arow > N - 1) arow = N - 1;
    const float* Arow = X + (size_t)arow * DIM;

    const float alpha = alpha_p[0];

    v8f acc[8];
    v8f zero = {};
#pragma unroll
    for (int t = 0; t < 8; ++t) acc[t] = zero;

    // K loop: 128 / 4 = 32 steps; 8 WMMA per step (one per 16-col tile).
    for (int kt = 0; kt < DIM; kt += 4) {
        const v2f a = *(const v2f*)(Arow + kt + halfSel * 2);   // 8B-aligned
        const v2f* Brow = Wp + ((kt >> 1) + halfSel) * DIM;     // this half's K pair
#pragma unroll
        for (int nt = 0; nt < 8; ++nt) {
            const v2f b = Brow[nt * 16 + l15];                  // ds_load_b64
            acc[nt] = __builtin_amdgcn_wmma_f32_16x16x4_f32(
                /*neg_a=*/false, a, /*neg_b=*/false, b,
                /*c_mod=*/(short)0, acc[nt],
                /*reuse_a=*/false, /*reuse_b=*/false);
        }
    }

    // Epilogue: bias + PReLU + store.
    // C/D layout: VGPR r -> M = r (lanes 0-15) / r+8 (lanes 16-31); N = l15.
    const int rowOff = halfSel * 8;
    if (rowBase + 16 <= N) {
        // Fast path: whole 16-row tile in range, no per-row guards.
#pragma unroll
        for (int nt = 0; nt < 8; ++nt) {
            const int col = nt * 16 + l15;
            const float bv = bias[col];
            float* Yt = Y + (size_t)(rowBase + rowOff) * DIM + col;
#pragma unroll
            for (int r = 0; r < 8; ++r) {
                float v = acc[nt][r] + bv;
                v = (v >= 0.0f) ? v : alpha * v;
                Yt[(size_t)r * DIM] = v;
            }
        }
    } else {
#pragma unroll
        for (int nt = 0; nt < 8; ++nt) {
            const int col = nt * 16 + l15;
            const float bv = bias[col];
#pragma unroll
            for (int r = 0; r < 8; ++r) {
                const int row = rowBase + r + rowOff;
                if (row < N) {
                    float v = acc[nt][r] + bv;
                    v = (v >= 0.0f) ? v : alpha * v;
                    Y[(size_t)row * DIM + col] = v;
                }
            }
        }
    }
}

// ------------------------------- Zero fill ---------------------------------
__global__ void zero_kernel(float* __restrict__ p, long long n)
{
    long long i = (long long)blockIdx.x * blockDim.x + threadIdx.x;
    if (i < n) p[i] = 0.0f;
}

// --------------------------------- SpMM ------------------------------------
// One wave per edge: lane L handles 4 consecutive floats (float4) of the
// 128-wide row. h and temp live in L2, so gather + atomic RMW stay on-chip.
__global__ __launch_bounds__(256) void spmm_atomic_kernel(
    const float* __restrict__ h,      // [N, 128]
    const int*   __restrict__ row,    // [E]
    const int*   __restrict__ col,    // [E]
    const float* __restrict__ val,    // [E]
    float* __restrict__ temp,         // [N, 128]
    int E)
{
    long long gid = (long long)blockIdx.x * blockDim.x + threadIdx.x;
    int e = (int)(gid >> 5);
    if (e >= E) return;
    const int lane = (int)(gid & 31);

    const int   r = row[e];
    const int   c = col[e];
    const float v = val[e];

    const float4 m = ((const float4*)(h + (size_t)c * DIM))[lane];
    float* tp = temp + (size_t)r * DIM + lane * 4;
    atomicAdd(tp + 0, v * m.x);
    atomicAdd(tp + 1, v * m.y);
    atomicAdd(tp + 2, v * m.z);
    atomicAdd(tp + 3, v * m.w);
}

// ------------------------------ Entry point --------------------------------
extern "C" void kernel_launch(void* const* d_in, const int* in_sizes, int n_in,
                              void* d_out, int out_size, void* d_ws, size_t ws_size,
                              hipStream_t stream)
{
    const float* AX        = (const float*)d_in[0];
    const int*   A_row     = (const int*)  d_in[1];
    const int*   A_col     = (const int*)  d_in[2];
    const float* A_val     = (const float*)d_in[3];
    const float* Wr_w      = (const float*)d_in[4];
    const float* Wr_b      = (const float*)d_in[5];
    const float* W_w       = (const float*)d_in[6];
    const float* W_b       = (const float*)d_in[7];
    const float* g_alpha   = (const float*)d_in[8];
    const float* act_alpha = (const float*)d_in[9];

    const int N = in_sizes[0] / DIM;   // 100000
    const int E = in_sizes[1];         // 1600000
    float* out = (float*)d_out;

    // Workspace: h then temp (each N*128 f32).
    float* h    = (float*)d_ws;
    float* temp = h + (size_t)N * DIM;

    const dim3 blk(256);
    const dim3 gemmGrid((N + 127) / 128);   // 8 waves x 16 rows per block

    // 1) h = prelu(AX @ Wr^T + Wr_b)
    gemm_prelu_wmma<<<gemmGrid, blk, 0, stream>>>(AX, Wr_w, Wr_b, g_alpha, h, N);

    // 2) temp = 0 (workspace is poisoned; must re-zero every call)
    const long long tn = (long long)N * DIM;
    zero_kernel<<<(unsigned)((tn + 255) / 256), blk, 0, stream>>>(temp, tn);

    // 3) temp[r] += val * h[c] over all edges (one wave per edge)
    const long long work = (long long)E * 32;
    spmm_atomic_kernel<<<(unsigned)((work + 255) / 256), blk, 0, stream>>>(
        h, A_row, A_col, A_val, temp, E);

    // 4) out = prelu(temp @ W^T + W_b)
    gemm_prelu_wmma<<<gemmGrid, blk, 0, stream>>>(temp, W_w, W_b, act_alpha, out, N);
}